// PILCAL_19095424598791
// MI455X (gfx1250) — compile-verified
//
#include <hip/hip_runtime.h>
#include <hip/hip_bf16.h>

#define TLEN 41
#define KLEN 28
#define BLK  256
#define L_MIN_F 1.0f
#define L_MAX_F 14.0f
#define INV_BW  (1.0f / 0.3f)

typedef __attribute__((ext_vector_type(4))) unsigned int u32x4;
typedef __attribute__((ext_vector_type(8))) int          i32x8;
typedef __attribute__((ext_vector_type(4))) int          i32x4;

typedef __attribute__((address_space(3))) char lds_char;

static __device__ __forceinline__ unsigned lds_addr_of(void* p) {
  // generic -> LDS addrspace -> 32-bit LDS byte offset
  return (unsigned)(unsigned long long)(lds_char*)p;
}

__global__ __launch_bounds__(BLK) void pil_kernel(
    const float* __restrict__ instock,
    const float* __restrict__ futured,
    const float* __restrict__ lead,
    float* __restrict__ out)
{
  __shared__ float s_f[BLK * TLEN];  // futured rows -> futured cumsum (in place)
  __shared__ float s_i[BLK * TLEN];  // instock rows -> outdate cummax (in place)

  const int tid = threadIdx.x;
  const int b0  = blockIdx.x * BLK;

#if __has_builtin(__builtin_amdgcn_tensor_load_to_lds)
  // --- CDNA5 Tensor Data Mover path: one wave DMAs the whole block tile ---
  if (tid < 32) {
    unsigned long long gaF =
        (unsigned long long)(const void*)(futured + (size_t)b0 * TLEN);
    unsigned long long gaI =
        (unsigned long long)(const void*)(instock + (size_t)b0 * TLEN);
    const unsigned ldsF = lds_addr_of((void*)s_f);
    const unsigned ldsI = lds_addr_of((void*)s_i);

    // D# group 1 (ISA 8.4): data_size=4B, tensor 41 x 256, stride 41, tile 41 x 256
    i32x8 g1;
    g1[0] = (2 << 16);        // [17:16] data_size = 2 -> 4 bytes
    g1[1] = (TLEN << 16);     // [79:48] tensor_dim0 = 41 (low 16 bits here)
    g1[2] = (BLK  << 16);     // [79:64]=dim0 hi=0, [95:80] tensor_dim1 lo = 256
    g1[3] = (TLEN << 16);     // [111:96]=dim1 hi=0, [127:112] tile_dim0 = 41
    g1[4] = BLK;              // [143:128] tile_dim1 = 256, [159:144] tile_dim2 = 0
    g1[5] = TLEN;             // [207:160] tensor_dim0_stride = 41 (low 32)
    g1[6] = 0;                // stride hi, tensor_dim1_stride lo
    g1[7] = 0;

    const i32x4 z4 = {0, 0, 0, 0};          // groups 2/3 unused (2D tensor)
    const i32x8 z8 = {0, 0, 0, 0, 0, 0, 0, 0};

    // D# group 0 (ISA 8.3): count=1, lds_addr, global_addr[56:0], type=2
    u32x4 g0;
    g0[0] = 1u;
    g0[1] = ldsF;
    g0[2] = (unsigned)(gaF & 0xFFFFFFFFull);
    g0[3] = (unsigned)((gaF >> 32) & 0x01FFFFFFull) | 0x80000000u;
    __builtin_amdgcn_tensor_load_to_lds(g0, g1, z4, z4, z8, 0);

    g0[1] = ldsI;
    g0[2] = (unsigned)(gaI & 0xFFFFFFFFull);
    g0[3] = (unsigned)((gaI >> 32) & 0x01FFFFFFull) | 0x80000000u;
    __builtin_amdgcn_tensor_load_to_lds(g0, g1, z4, z4, z8, 0);

#if __has_builtin(__builtin_amdgcn_s_wait_tensorcnt)
    __builtin_amdgcn_s_wait_tensorcnt(0);
#else
    asm volatile("s_wait_tensorcnt 0" ::: "memory");
#endif
  }
  __syncthreads();
#else
  // --- fallback: coalesced manual staging ---
  for (int i = tid; i < BLK * TLEN; i += BLK) {
    s_f[i] = futured[(size_t)b0 * TLEN + i];
    s_i[i] = instock[(size_t)b0 * TLEN + i];
  }
  __syncthreads();
#endif

  // Per-thread row pointers; stride 41 floats, gcd(41,64)=1 -> conflict-free.
  float* fr = &s_f[tid * TLEN];
  float* ir = &s_i[tid * TLEN];

  // cumsum(futured), cumsum(instock), running max of diff (head relu'd)
  float rf = 0.f, ri = 0.f, om = 0.f;
#pragma unroll
  for (int t = 0; t < TLEN; ++t) {
    rf += fr[t];
    ri += ir[t];
    fr[t] = rf;
    const float d = ri - rf;
    om = (t == 0) ? fmaxf(d, 0.f) : fmaxf(om, d);
    ir[t] = om;
  }
  const float ic_last = ri;

  float L = lead[b0 + tid] * (L_MAX_F - L_MIN_F) + L_MIN_F;
  L = fminf(fmaxf(L, L_MIN_F), L_MAX_F);

  // 9-tap Gaussian window around x ~= L (x = t+1-k). Taps beyond |x-L|>=3
  // are < 2e-22 relative -> identical to the full sum in fp32.
  const int x0 = (int)floorf(L) - 4;
  float w[9];
#pragma unroll
  for (int i = 0; i < 9; ++i) {
    const float d = ((float)(x0 + i) - L) * INV_BW;
    w[i] = __expf(-0.5f * d * d);
  }

  float res[KLEN];
#pragma unroll
  for (int k = 0; k < KLEN; ++k) {
    float s = 0.f, wd = 0.f, wo = 0.f;
    const int tb = x0 + k - 1;  // t for tap i=0  (t+1 = x0+i+k)
#pragma unroll
    for (int i = 0; i < 9; ++i) {
      const int t = tb + i;
      if (t >= 0 && t < TLEN) {
        const float ww = w[i];
        s  += ww;
        wd += ww * fr[t];
        wo += ww * ir[t];
      }
    }
    res[k] = ic_last - (wd + wo) / s;
  }

  // 28 floats per row; row base = b*112 bytes, 16B aligned -> 7x float4.
  float4* op = (float4*)(out + (size_t)(b0 + tid) * KLEN);
#pragma unroll
  for (int j = 0; j < KLEN / 4; ++j) {
    float4 v;
    v.x = res[4 * j + 0];
    v.y = res[4 * j + 1];
    v.z = res[4 * j + 2];
    v.w = res[4 * j + 3];
    op[j] = v;
  }
}

extern "C" void kernel_launch(void* const* d_in, const int* in_sizes, int n_in,
                              void* d_out, int out_size, void* d_ws, size_t ws_size,
                              hipStream_t stream) {
  const float* instock = (const float*)d_in[0];
  const float* futured = (const float*)d_in[1];
  const float* lead    = (const float*)d_in[2];
  float* out = (float*)d_out;

  const int batch = in_sizes[0] / TLEN;  // 65536
  dim3 grid(batch / BLK), block(BLK);
  hipLaunchKernelGGL(pil_kernel, grid, block, 0, stream,
                     instock, futured, lead, out);
}